// ModulatedChunks_64871186039380
// MI455X (gfx1250) — compile-verified
//
#include <hip/hip_runtime.h>
#include <hip/hip_bf16.h>

// ---------------------------------------------------------------------------
// ModulatedChunks for MI455X (gfx1250, wave32).
// Algorithmic rewrite: chunk k of window w reads rows s..s+3 with s = w + 4k,
// so all chunk-level quantities depend only on (b, s), s in [0, 1021).
// GEMMs run on V_WMMA_F32_16X16X4_F32 (fp32 matrix path, exact precision for
// the argmax-sensitive stages), register-blocked 64Mx32N per wave so the
// inner loop issues ~1 VMEM op per WMMA instead of 3-6.
// ---------------------------------------------------------------------------

typedef __attribute__((ext_vector_type(2))) float v2f;
typedef __attribute__((ext_vector_type(8))) float v8f;

#define Bb   8
#define Tt   1024
#define Cc   512
#define Qq   768
#define Ll   24
#define NWIN 993          // T - 32 + 1
#define NS   1021         // T - 4 + 1 (distinct chunk starts)
#define MROWS (Bb*NS)     // 8168 pooled rows

// ---------------------------------------------------------------------------
// K1: e1 = relu(q @ Wq1 + b1), e2 = relu(q @ Wq2 + b2). One block per (b,l).
// ---------------------------------------------------------------------------
__global__ void mc_queries(const float* __restrict__ q,
                           const float* __restrict__ Wq1, const float* __restrict__ bq1,
                           const float* __restrict__ Wq2, const float* __restrict__ bq2,
                           float* __restrict__ e1, float* __restrict__ e2)
{
    __shared__ float qs[Qq];
    const int bl = blockIdx.x;                 // 0 .. B*L-1
    const float* qrow = q + (size_t)bl * Qq;
    for (int i = threadIdx.x; i < Qq; i += 256) qs[i] = qrow[i];
    __syncthreads();
    for (int c = threadIdx.x; c < Cc; c += 256) {
        float s1 = bq1[c], s2 = bq2[c];
        for (int k = 0; k < Qq; ++k) {
            float qv = qs[k];
            s1 = fmaf(qv, Wq1[k * Cc + c], s1);
            s2 = fmaf(qv, Wq2[k * Cc + c], s2);
        }
        e1[(size_t)bl * Cc + c] = fmaxf(s1, 0.f);
        e2[(size_t)bl * Cc + c] = fmaxf(s2, 0.f);
    }
}

// ---------------------------------------------------------------------------
// Blocked GEMM: Out[m,n] = relu(A[m,:] @ W[:,n] + bias[n]), N = 512, K = 512.
// One wave computes a 64x32 tile as 4x2 WMMA 16x16 sub-tiles.
// Fragment layouts (ISA 7.12.2), h = lane>>4:
//   A 16x4 f32: lane holds M=lane&15; element j holds K = j + 2h
//   B 4x16 f32: lane holds N=lane&15; element j holds K = j + 2h
//   C 16x16 f32: element r at lane is row r + 8h, col lane&15
// Per K-step of 4: 4 x b64 (A) + 4 x b32 (B) loads feed 8 WMMAs.
// Rows >= Mrows: loads clamped to row 0, stores guarded.
// ---------------------------------------------------------------------------
__global__ void mc_gemm_blocked(const float* __restrict__ A, int Mrows,
                                const float* __restrict__ W,   // 512 x 512 (k-major)
                                const float* __restrict__ bias,
                                float* __restrict__ Out)
{
    const int wave = threadIdx.x >> 5;
    const int lane = threadIdx.x & 31;
    const int tile = blockIdx.x * 8 + wave;
    const int nt   = tile & 15;            // 16 n-tiles of 32 cols
    const int mt   = tile >> 4;            // 64-row m-tiles
    const int r    = lane & 15;
    const int h    = lane >> 4;
    const int h2   = h * 2;

    const float* arow[4];
#pragma unroll
    for (int i = 0; i < 4; ++i) {
        int m = mt * 64 + i * 16 + r;
        arow[i] = A + (size_t)((m < Mrows) ? m : 0) * Cc;
    }
    const int n0 = nt * 32 + r;
    const int n1 = n0 + 16;

    v8f acc[4][2] = {};
    for (int kb = 0; kb < Cc; kb += 4) {
        const int k0 = kb + h2;                       // even -> 8B aligned
        const float* w0 = W + (size_t)k0 * Cc;
        v2f b0, b1;
        b0.x = w0[n0];      b0.y = w0[Cc + n0];
        b1.x = w0[n1];      b1.y = w0[Cc + n1];
#pragma unroll
        for (int i = 0; i < 4; ++i) {
            v2f a = *(const v2f*)(arow[i] + k0);
            acc[i][0] = __builtin_amdgcn_wmma_f32_16x16x4_f32(false, a, false, b0,
                                                              (short)0, acc[i][0], false, false);
            acc[i][1] = __builtin_amdgcn_wmma_f32_16x16x4_f32(false, a, false, b1,
                                                              (short)0, acc[i][1], false, false);
        }
    }

    const float bn0 = bias[n0], bn1 = bias[n1];
#pragma unroll
    for (int i = 0; i < 4; ++i) {
        const int mbase = mt * 64 + i * 16 + 8 * h;
#pragma unroll
        for (int rr = 0; rr < 8; ++rr) {
            const int m = mbase + rr;
            if (m < Mrows) {
                Out[(size_t)m * Cc + n0] = fmaxf(acc[i][0][rr] + bn0, 0.f);
                Out[(size_t)m * Cc + n1] = fmaxf(acc[i][1][rr] + bn1, 0.f);
            }
        }
    }
}

// ---------------------------------------------------------------------------
// K3: labels[b,t] = argmax_l dot(v[b,t,:], e1[b,l,:]).  One wave per (b,t).
// ---------------------------------------------------------------------------
__global__ void mc_labels(const float* __restrict__ V,
                          const float* __restrict__ e1,
                          int* __restrict__ labels)
{
    const int wave = threadIdx.x >> 5;
    const int lane = threadIdx.x & 31;
    const int bt = blockIdx.x * 8 + wave;          // 8192 rows
    const int b  = bt >> 10;
    const float* vrow = V + (size_t)bt * Cc;

    float vr[16];
#pragma unroll
    for (int i = 0; i < 16; ++i) vr[i] = vrow[lane + 32 * i];

    float best = -3.0e38f;
    int bestl = 0;
    for (int l = 0; l < Ll; ++l) {
        const float* er = e1 + (size_t)(b * Ll + l) * Cc;
        float acc = 0.f;
#pragma unroll
        for (int i = 0; i < 16; ++i) acc = fmaf(vr[i], er[lane + 32 * i], acc);
#pragma unroll
        for (int off = 16; off > 0; off >>= 1) acc += __shfl_xor(acc, off, 32);
        if (acc > best) { best = acc; bestl = l; }  // strict > => first max
    }
    if (lane == 0) labels[bt] = bestl;
}

// ---------------------------------------------------------------------------
// K4: Mo[b,s] = mode of labels[b, s..s+3]; ties -> smallest label
// (== argmax over one-hot counts).
// ---------------------------------------------------------------------------
__global__ void mc_mode(const int* __restrict__ labels, int* __restrict__ Mo)
{
    const int idx = blockIdx.x * 256 + threadIdx.x;
    if (idx >= MROWS) return;
    const int b = idx / NS, s = idx % NS;
    const int* lb = labels + b * Tt + s;
    int la[4] = { lb[0], lb[1], lb[2], lb[3] };
    int bestLab = 0, bestCnt = -1;
#pragma unroll
    for (int i = 0; i < 4; ++i) {
        int c = 0;
#pragma unroll
        for (int j = 0; j < 4; ++j) c += (la[j] == la[i]) ? 1 : 0;
        if (c > bestCnt || (c == bestCnt && la[i] < bestLab)) { bestCnt = c; bestLab = la[i]; }
    }
    Mo[idx] = bestLab;
}

// ---------------------------------------------------------------------------
// K4b: S[b,s,:] = mean(v[b,s..s+3,:]).  float4 per thread; v is L2-resident.
// ---------------------------------------------------------------------------
__global__ void mc_pool4(const float* __restrict__ V, float* __restrict__ S)
{
    const int idx = blockIdx.x * 256 + threadIdx.x;      // (m, c/4)
    if (idx >= MROWS * (Cc / 4)) return;
    const int m  = idx >> 7;                             // / (Cc/4) = 128
    const int c4 = (idx & 127) << 2;
    const int b = m / NS, s = m % NS;
    const float* p = V + (size_t)(b * Tt + s) * Cc + c4;
    float4 x0 = *(const float4*)(p);
    float4 x1 = *(const float4*)(p + Cc);
    float4 x2 = *(const float4*)(p + 2 * Cc);
    float4 x3 = *(const float4*)(p + 3 * Cc);
    float4 o;
    o.x = 0.25f * (x0.x + x1.x + x2.x + x3.x);
    o.y = 0.25f * (x0.y + x1.y + x2.y + x3.y);
    o.z = 0.25f * (x0.z + x1.z + x2.z + x3.z);
    o.w = 0.25f * (x0.w + x1.w + x2.w + x3.w);
    *(float4*)(S + (size_t)m * Cc + c4) = o;
}

// ---------------------------------------------------------------------------
// K6: P[b,s,j] = sum_c e2[b,Mo[b,s],c]*G[b,s,c]*W_p[c,j] + b_p[j];
// scatter to out[b,j,w,k] for all (w,k) with w + 4k == s.  One wave per (b,s).
// ---------------------------------------------------------------------------
__global__ void mc_final(const float* __restrict__ G,
                         const float* __restrict__ e2,
                         const int* __restrict__ Mo,
                         const float* __restrict__ Wp,   // 512 x 2
                         const float* __restrict__ bp,   // 2
                         float* __restrict__ out)        // (8,2,993,8)
{
    const int wave = threadIdx.x >> 5;
    const int lane = threadIdx.x & 31;
    const int idx = blockIdx.x * 8 + wave;         // 0 .. 8167 (grid padded)
    if (idx >= MROWS) return;
    const int b = idx / NS, s = idx % NS;
    const float* grow = G + (size_t)idx * Cc;
    const float* erow = e2 + (size_t)(b * Ll + Mo[idx]) * Cc;

    float p0 = 0.f, p1 = 0.f;
#pragma unroll
    for (int i = 0; i < 16; ++i) {
        const int c = lane + 32 * i;
        const float m = grow[c] * erow[c];
        p0 = fmaf(m, Wp[c * 2 + 0], p0);
        p1 = fmaf(m, Wp[c * 2 + 1], p1);
    }
#pragma unroll
    for (int off = 16; off > 0; off >>= 1) {
        p0 += __shfl_xor(p0, off, 32);
        p1 += __shfl_xor(p1, off, 32);
    }
    if (lane == 0) {
        p0 += bp[0];
        p1 += bp[1];
#pragma unroll
        for (int k = 0; k < 8; ++k) {
            const int w = s - 4 * k;
            if (w >= 0 && w < NWIN) {
                out[((size_t)(b * 2 + 0) * NWIN + w) * 8 + k] = p0;
                out[((size_t)(b * 2 + 1) * NWIN + w) * 8 + k] = p1;
            }
        }
    }
}

// ---------------------------------------------------------------------------
extern "C" void kernel_launch(void* const* d_in, const int* in_sizes, int n_in,
                              void* d_out, int out_size, void* d_ws, size_t ws_size,
                              hipStream_t stream) {
    const float* vis  = (const float*)d_in[0];   // (8,1024,512)
    const float* qf   = (const float*)d_in[1];   // (8,24,768)
    const float* Wv1  = (const float*)d_in[2];
    const float* bv1  = (const float*)d_in[3];
    const float* Wv2  = (const float*)d_in[4];
    const float* bv2  = (const float*)d_in[5];
    const float* Wq1  = (const float*)d_in[6];
    const float* bq1  = (const float*)d_in[7];
    const float* Wq2  = (const float*)d_in[8];
    const float* bq2  = (const float*)d_in[9];
    const float* Wp   = (const float*)d_in[10];
    const float* bp   = (const float*)d_in[11];
    float* out = (float*)d_out;

    // Workspace layout, ~34.4 MB.  v dies after mc_pool4, so G aliases it.
    float* ws = (float*)d_ws;
    float* v   = ws;                               // 8192*512  (later reused as G)
    float* G   = v;
    float* S   = v  + (size_t)8192 * Cc;           // 8168*512 (padded region)
    float* e1  = S  + (size_t)8192 * Cc;           // 192*512
    float* e2  = e1 + (size_t)Bb * Ll * Cc;        // 192*512
    int* labels = (int*)(e2 + (size_t)Bb * Ll * Cc);  // 8192
    int* Mo     = labels + Bb * Tt;                   // 8168

    // K1: query embeddings
    mc_queries<<<Bb * Ll, 256, 0, stream>>>(qf, Wq1, bq1, Wq2, bq2, e1, e2);

    // K2: v = relu(vis @ Wv1 + bv1); (8192/64)*(512/32) = 2048 wave-tiles
    mc_gemm_blocked<<<2048 / 8, 256, 0, stream>>>(vis, Bb * Tt, Wv1, bv1, v);

    // K3: labels (reads v, e1)
    mc_labels<<<(Bb * Tt) / 8, 256, 0, stream>>>(v, e1, labels);

    // K4: per-start mode
    mc_mode<<<(MROWS + 255) / 256, 256, 0, stream>>>(labels, Mo);

    // K4b: sliding 4-row mean (reads v, writes S) -- v dead afterwards
    mc_pool4<<<(MROWS * (Cc / 4) + 255) / 256, 256, 0, stream>>>(v, S);

    // K5: G = relu(S @ Wv2 + bv2); M padded 8168 -> 8192 -> 2048 wave-tiles
    mc_gemm_blocked<<<2048 / 8, 256, 0, stream>>>(S, MROWS, Wv2, bv2, G);

    // K6: modulated projection + scatter
    mc_final<<<(MROWS + 7) / 8, 256, 0, stream>>>(G, e2, Mo, Wp, bp, out);
}